// SE_88407606821537
// MI455X (gfx1250) — compile-verified
//
#include <hip/hip_runtime.h>
#include <hip/hip_bf16.h>
#include <math.h>

// -------- problem constants (from reference setup) --------
#define N_BATCH 64
#define C_CH    256
#define S_CH    16
#define HW      (56 * 56)          // 3136 floats per (n,c) plane
#define HW4     (HW / 4)           // 784 float4 per plane
#define PLANES  (N_BATCH * C_CH)   // 16384

typedef __attribute__((ext_vector_type(2))) float v2f;
typedef __attribute__((ext_vector_type(4))) float v4f;
typedef __attribute__((ext_vector_type(8))) float v8f;

// ============================================================
// Kernel 1: global average pool.  One 256-thread block (8 waves)
// per (n,c) plane. Regular-temporal loads: pull x into L2 so the
// scale pass can hit it there (x = 205MB vs 192MB L2).
// ============================================================
__global__ void se_pool_kernel(const float* __restrict__ x,
                               float* __restrict__ s /* [N,C] */) {
    const int plane = blockIdx.x;                       // n*C + c
    const v4f* xp = (const v4f*)(x + (size_t)plane * HW);

    float sum = 0.0f;
    for (int i = threadIdx.x; i < HW4; i += blockDim.x) {
        v4f v = xp[i];
        sum += (v.x + v.y) + (v.z + v.w);
    }
    // wave32 tree reduction
    #pragma unroll
    for (int off = 16; off > 0; off >>= 1)
        sum += __shfl_xor(sum, off, 32);

    __shared__ float wsum[8];
    const int wave = threadIdx.x >> 5;
    const int lane = threadIdx.x & 31;
    if (lane == 0) wsum[wave] = sum;
    __syncthreads();
    if (threadIdx.x == 0) {
        float t = 0.0f;
        #pragma unroll
        for (int w = 0; w < 8; ++w) t += wsum[w];
        s[plane] = t * (1.0f / (float)HW);
    }
}

// ============================================================
// Kernel 2: excitation MLP via V_WMMA_F32_16X16X4_F32.
// h = swish(s @ w1^T + b1)      [64,16]
// g = sigmoid(h @ w2^T + b2)    [64,256]
// 1 block, 128 threads = 4 waves; wave m owns rows 16m..16m+15.
// EXEC all-1s as WMMA requires.
//
// Fragment layouts (ISA 7.12.2, wave32):
//   A (16x4 f32):  lanes 0-15: M=lane,    VGPR0=K0, VGPR1=K1
//                  lanes16-31: M=lane-16, VGPR0=K2, VGPR1=K3
//   B (4x16 f32):  lanes 0-15: N=lane,    VGPR0=K0, VGPR1=K1
//                  lanes16-31: N=lane-16, VGPR0=K2, VGPR1=K3
//   C/D (16x16):   VGPR v: M = v + 8*(lane>=16), N = lane%16
// ============================================================
__global__ void se_mlp_kernel(const float* __restrict__ s,   // [64,256]
                              const float* __restrict__ w1,  // [16,256]
                              const float* __restrict__ b1,  // [16]
                              const float* __restrict__ w2,  // [256,16]
                              const float* __restrict__ b2,  // [256]
                              float* __restrict__ g) {       // [64,256]
    const int lane  = threadIdx.x & 31;
    const int wave  = threadIdx.x >> 5;   // M-tile 0..3
    const int mBase = wave * 16;
    const int half  = lane >> 4;          // 0 or 1
    const int l16   = lane & 15;

    __shared__ float hLds[N_BATCH * S_CH]; // [64,16]

    // ---- GEMM1: D[16,16] per wave (N = S_CH = 16 exactly) ----
    v8f acc = {};
    for (int k0 = 0; k0 < C_CH; k0 += 4) {
        const int k = k0 + 2 * half;
        // A: s[mBase + l16, k], s[mBase + l16, k+1]
        const float* arow = s + (size_t)(mBase + l16) * C_CH + k;
        v2f a; a.x = arow[0]; a.y = arow[1];
        // B[k][n] = w1[n][k]  (w1 is [S,C] row-major)
        const float* brow = w1 + (size_t)l16 * C_CH + k;
        v2f b; b.x = brow[0]; b.y = brow[1];
        acc = __builtin_amdgcn_wmma_f32_16x16x4_f32(
            /*neg_a=*/false, a, /*neg_b=*/false, b,
            /*c_mod=*/(short)0, acc, /*reuse_a=*/false, /*reuse_b=*/false);
    }
    // bias + swish, stage h in LDS
    {
        const float bias1 = b1[l16];
        #pragma unroll
        for (int v = 0; v < 8; ++v) {
            const int m = mBase + v + half * 8;
            float hv = acc[v] + bias1;
            hv = hv * (1.0f / (1.0f + __expf(-hv)));   // swish = h*sigmoid(h)
            hLds[m * S_CH + l16] = hv;
        }
    }
    __syncthreads();

    // ---- GEMM2: wave m covers rows mBase..mBase+15, all 16 N-tiles ----
    for (int nt = 0; nt < C_CH / 16; ++nt) {
        const int nCol  = nt * 16 + l16;
        const float bias2 = b2[nCol];
        v8f acc2 = {};
        #pragma unroll
        for (int k0 = 0; k0 < S_CH; k0 += 4) {
            const int k = k0 + 2 * half;
            v2f a;
            a.x = hLds[(mBase + l16) * S_CH + k];
            a.y = hLds[(mBase + l16) * S_CH + k + 1];
            // B[k][n] = w2[n][k]  (w2 is [C,S] row-major)
            const float* brow = w2 + (size_t)nCol * S_CH + k;
            v2f b; b.x = brow[0]; b.y = brow[1];
            acc2 = __builtin_amdgcn_wmma_f32_16x16x4_f32(
                false, a, false, b, (short)0, acc2, false, false);
        }
        #pragma unroll
        for (int v = 0; v < 8; ++v) {
            const int m = mBase + v + half * 8;
            float pre  = acc2[v] + bias2;
            g[m * C_CH + nCol] = 1.0f / (1.0f + __expf(-pre));  // sigmoid
        }
    }
}

// ============================================================
// Kernel 3: out = x * g[plane].  Reverse plane order so the most
// recently pooled (hottest-in-L2) planes are consumed first.
// NT load of x (last use) + NT store of out (never re-read) to
// avoid evicting still-needed x lines from the 192MB L2.
// ============================================================
__global__ void se_scale_kernel(const float* __restrict__ x,
                                const float* __restrict__ g,
                                float* __restrict__ out) {
    const int plane = (PLANES - 1) - blockIdx.x;       // reverse order
    const float gate = g[plane];
    const v4f* xp = (const v4f*)(x  + (size_t)plane * HW);
    v4f*       op = (v4f*)(out + (size_t)plane * HW);

    for (int i = threadIdx.x; i < HW4; i += blockDim.x) {
        v4f v = __builtin_nontemporal_load(&xp[i]);
        v *= gate;
        __builtin_nontemporal_store(v, &op[i]);
    }
}

// ============================================================
extern "C" void kernel_launch(void* const* d_in, const int* in_sizes, int n_in,
                              void* d_out, int out_size, void* d_ws, size_t ws_size,
                              hipStream_t stream) {
    const float* x  = (const float*)d_in[0];   // [64,256,56,56]
    const float* w1 = (const float*)d_in[1];   // [16,256]
    const float* b1 = (const float*)d_in[2];   // [16]
    const float* w2 = (const float*)d_in[3];   // [256,16]
    const float* b2 = (const float*)d_in[4];   // [256]
    float* out = (float*)d_out;

    float* s = (float*)d_ws;                   // [64,256] = 64KB
    float* g = s + PLANES;                     // [64,256] = 64KB

    se_pool_kernel<<<PLANES, 256, 0, stream>>>(x, s);
    se_mlp_kernel<<<1, 128, 0, stream>>>(s, w1, b1, w2, b2, g);
    se_scale_kernel<<<PLANES, 256, 0, stream>>>(x, g, out);
}